// GridMask_78554951844113
// MI455X (gfx1250) — compile-verified
//
#include <hip/hip_runtime.h>

// GridMask: out[n,c,s,h,w] = x * ((h+OFF_H-st_h[n,s]) % d[n] < ceil(d/2)
//                               || (w+OFF_W-st_w[n,s]) % d[n] < ceil(d/2))
// Pure streaming multiply: 402 MB in + 402 MB out @ 23.3 TB/s ~ 35 us floor.
// AI ~ 0.125 FLOP/byte -> no matrix math, WMMA inapplicable by roofline.
// Strategy: separable mask (row OR col) precomputed once per block into LDS,
// then 16 nontemporal b128 loads in flight per lane (8 KB/wave outstanding)
// to cover HBM latency, multiply, nontemporal b128 stores.

typedef float vfloat4 __attribute__((ext_vector_type(4)));

namespace {
constexpr int Nn = 8, Cc = 3, Ss = 16, Hh = 512, Ww = 512;
constexpr int OFFH = 106, OFFW = 106;   // (ceil(512*sqrt(2)) - 512) / 2 = (725-512)/2
constexpr int TILE_ROWS = 32;
constexpr int W4 = Ww / 4;              // 128 float4 per row
constexpr int THREADS = 256;            // 8 wave32s
constexpr int F4_PER_THREAD = (TILE_ROWS * W4) / THREADS;  // 16
}

__global__ __launch_bounds__(THREADS)
void gridmask_kernel(const float* __restrict__ x,
                     const int*   __restrict__ d,
                     const int*   __restrict__ st_h,
                     const int*   __restrict__ st_w,
                     float*       __restrict__ out)
{
    __shared__ __align__(16) float colMask[Ww];   // 1.0 where column stripe hits
    __shared__            float rowHit[TILE_ROWS];

    const int b  = blockIdx.x;
    const int ht = b & 15;            // 16 row-tiles of 32 rows
    const int s  = (b >> 4) & 15;
    const int nc = b >> 8;            // n*3 + c
    const int c  = nc % 3;
    const int n  = nc / 3;

    // Per-(n,s) stripe parameters (uniform across block -> scalar path).
    const int dn = d[n];
    const int l  = (dn + 1) >> 1;                 // ceil(d * 0.5) exactly
    const int sh = st_h[n * Ss + s] % dn;
    const int sw = st_w[n * Ss + s] % dn;

    const int t = threadIdx.x;

    // Build separable mask factors once per block (544 int-mods amortized
    // over 128 KB of streamed data).
    for (int i = t; i < Ww; i += THREADS) {
        int v = (i + OFFW - sw) % dn; if (v < 0) v += dn;
        colMask[i] = (v < l) ? 1.0f : 0.0f;
    }
    if (t < TILE_ROWS) {
        int row = ht * TILE_ROWS + t;
        int v = (row + OFFH - sh) % dn; if (v < 0) v += dn;
        rowHit[t] = (v < l) ? 1.0f : 0.0f;
    }
    __syncthreads();

    // This block's 32x512 tile of plane (n,c,s), as float4.
    const size_t base4 =
        ((size_t)((n * Cc + c) * Ss + s) * Hh + (size_t)ht * TILE_ROWS) * W4;
    const vfloat4* __restrict__ xin = reinterpret_cast<const vfloat4*>(x) + base4;
    vfloat4*       __restrict__ xo  = reinterpret_cast<vfloat4*>(out)   + base4;
    const vfloat4* cm4 = reinterpret_cast<const vfloat4*>(colMask);

    // Issue all 16 b128 loads first (8 KB in flight per wave), nontemporal:
    // the 805 MB stream must not thrash the 192 MB L2.
    vfloat4 xv[F4_PER_THREAD];
#pragma unroll
    for (int k = 0; k < F4_PER_THREAD; ++k) {
        xv[k] = __builtin_nontemporal_load(xin + (t + k * THREADS));
    }

#pragma unroll
    for (int k = 0; k < F4_PER_THREAD; ++k) {
        const int j  = t + k * THREADS;
        const int r  = j >> 7;          // wave-uniform row within tile
        const int c4 = j & (W4 - 1);    // conflict-free LDS read
        const float   rh = rowHit[r];
        const vfloat4 cm = cm4[c4];
        vfloat4 o;
        o.x = xv[k].x * fmaxf(rh, cm.x);   // rh,cm in {0,1}: OR == max
        o.y = xv[k].y * fmaxf(rh, cm.y);
        o.z = xv[k].z * fmaxf(rh, cm.z);
        o.w = xv[k].w * fmaxf(rh, cm.w);
        __builtin_nontemporal_store(o, xo + j);
    }
}

extern "C" void kernel_launch(void* const* d_in, const int* in_sizes, int n_in,
                              void* d_out, int out_size, void* d_ws, size_t ws_size,
                              hipStream_t stream) {
    (void)in_sizes; (void)n_in; (void)out_size; (void)d_ws; (void)ws_size;
    const float* x    = (const float*)d_in[0];
    const int*   d    = (const int*)  d_in[1];
    const int*   st_h = (const int*)  d_in[2];
    const int*   st_w = (const int*)  d_in[3];
    float*       out  = (float*)d_out;

    const int blocks = Nn * Cc * Ss * (Hh / TILE_ROWS);   // 6144
    gridmask_kernel<<<blocks, THREADS, 0, stream>>>(x, d, st_h, st_w, out);
}